// QNetwork_46256797778567
// MI455X (gfx1250) — compile-verified
//
#include <hip/hip_runtime.h>
#include <hip/hip_bf16.h>
#include <stdint.h>

// ---------------------------------------------------------------------------
// VQ-VAE forward for MI455X (gfx1250, wave32).
// WMMA f32 16x16x4 used for:  (1) VQ distance GEMM  q[65536x64] @ emb^T[64x512]
//                             (2) Q-head layer 1    flat[1024x4096] @ lw1[4096x512]
// Q-head layer 1 stages its B-matrix through LDS with CDNA5 async
// global->LDS copies (ASYNCcnt), double-buffered, shared by 8 waves/block.
// Convs/deconvs are direct VALU kernels (bandwidth-bound at this size).
// ---------------------------------------------------------------------------

typedef __attribute__((ext_vector_type(2))) float v2f;
typedef __attribute__((ext_vector_type(8))) float v8f;

#define NB      1024
#define KCODES  512
#define NACT    18
#define NPOS    (NB * 64)            // 65536 spatial positions
#define VQ_ELEMS (NPOS * 64)         // 4,194,304
#define REC_ELEMS (NB * 4 * 84 * 84) // 28,901,376

// ------------------------- direct conv (+ReLU) -----------------------------
template<int CIN, int HIN, int COUT, int HOUT, int KS, int STR, bool POSMAJOR>
__global__ __launch_bounds__(256) void conv_relu_kernel(
    const float* __restrict__ in, const float* __restrict__ w,
    const float* __restrict__ bias, float* __restrict__ out, float in_scale)
{
    int t = blockIdx.x * 256 + threadIdx.x;
    const int total = NB * COUT * HOUT * HOUT;
    if (t >= total) return;
    int x  = t % HOUT;
    int y  = (t / HOUT) % HOUT;
    int co = (t / (HOUT * HOUT)) % COUT;
    int n  = t / (HOUT * HOUT * COUT);

    const float* ip = in + (long)n * CIN * HIN * HIN;
    const float* wp = w + co * CIN * KS * KS;
    float s = 0.f;
    for (int ci = 0; ci < CIN; ++ci) {
        const float* ipc = ip + ci * HIN * HIN;
        const float* wpc = wp + ci * KS * KS;
        #pragma unroll
        for (int ky = 0; ky < KS; ++ky)
            #pragma unroll
            for (int kx = 0; kx < KS; ++kx)
                s += ipc[(y * STR + ky) * HIN + (x * STR + kx)] * wpc[ky * KS + kx];
    }
    float acc = s * in_scale + bias[co];
    acc = acc > 0.f ? acc : 0.f;
    if (POSMAJOR) {
        out[((n * HOUT * HOUT) + y * HOUT + x) * COUT + co] = acc;
    } else {
        out[t] = acc;
    }
}

// ------------------------- direct deconv (+ReLU) ---------------------------
// reference: conv with lhs_dilation=STR, pad KS-1, weights OIHW (no flip)
template<int CIN, int HIN, int COUT, int KS, int STR>
__global__ __launch_bounds__(256) void deconv_relu_kernel(
    const float* __restrict__ in, const float* __restrict__ w,
    const float* __restrict__ bias, float* __restrict__ out)
{
    const int HOUT = (HIN - 1) * STR + KS;
    int t = blockIdx.x * 256 + threadIdx.x;
    const int total = NB * COUT * HOUT * HOUT;
    if (t >= total) return;
    int ox = t % HOUT;
    int oy = (t / HOUT) % HOUT;
    int co = (t / (HOUT * HOUT)) % COUT;
    int n  = t / (HOUT * HOUT * COUT);

    float acc = bias[co];
    #pragma unroll
    for (int ky = 0; ky < KS; ++ky) {
        int p = oy + ky - (KS - 1);
        if (p < 0 || p > (HIN - 1) * STR || (p % STR)) continue;
        int iy = p / STR;
        #pragma unroll
        for (int kx = 0; kx < KS; ++kx) {
            int qx = ox + kx - (KS - 1);
            if (qx < 0 || qx > (HIN - 1) * STR || (qx % STR)) continue;
            int ix = qx / STR;
            const float* ip = in + ((long)n * CIN) * HIN * HIN + iy * HIN + ix;
            const float* wp = w + (co * CIN) * KS * KS + ky * KS + kx;
            for (int ci = 0; ci < CIN; ++ci)
                acc += ip[ci * HIN * HIN] * wp[ci * KS * KS];
        }
    }
    out[t] = acc > 0.f ? acc : 0.f;
}

// ------------ deconv3 fused with sigmoid + reconstruction loss -------------
__global__ __launch_bounds__(256) void deconv3_loss_kernel(
    const float* __restrict__ in,   // d2 [1024,32,21,21]
    const float* __restrict__ w,    // dw3 [4,32,4,4]
    const float* __restrict__ bias, // db3 [4]
    const float* __restrict__ xraw, // x [1024,4,84,84] (unscaled)
    float* __restrict__ partial)    // per-block loss partial
{
    const int CIN = 32, HIN = 21, KS = 4, STR = 4, HOUT = 84;
    int t = blockIdx.x * 256 + threadIdx.x;
    float val = 0.f;
    if (t < REC_ELEMS) {
        int ox = t % HOUT;
        int oy = (t / HOUT) % HOUT;
        int co = (t / (HOUT * HOUT)) % 4;
        int n  = t / (HOUT * HOUT * 4);
        float acc = bias[co];
        #pragma unroll
        for (int ky = 0; ky < KS; ++ky) {
            int p = oy + ky - (KS - 1);
            if (p < 0 || p > (HIN - 1) * STR || (p % STR)) continue;
            int iy = p / STR;
            #pragma unroll
            for (int kx = 0; kx < KS; ++kx) {
                int qx = ox + kx - (KS - 1);
                if (qx < 0 || qx > (HIN - 1) * STR || (qx % STR)) continue;
                int ix = qx / STR;
                const float* ip = in + ((long)n * CIN) * HIN * HIN + iy * HIN + ix;
                const float* wp = w + (co * CIN) * KS * KS + ky * KS + kx;
                for (int ci = 0; ci < CIN; ++ci)
                    acc += ip[ci * HIN * HIN] * wp[ci * KS * KS];
            }
        }
        float dec = 1.f / (1.f + __expf(-acc));
        float diff = dec - xraw[t] * (1.0f / 255.0f);
        val = diff * diff;
    }
    __shared__ float sred[256];
    sred[threadIdx.x] = val;
    __syncthreads();
    for (int s = 128; s; s >>= 1) {
        if (threadIdx.x < s) sred[threadIdx.x] += sred[threadIdx.x + s];
        __syncthreads();
    }
    if (threadIdx.x == 0) partial[blockIdx.x] = sred[0];
}

// ------------------------- codebook row norms ------------------------------
__global__ void enorm_kernel(const float* __restrict__ emb, float* __restrict__ enorm)
{
    int k = blockIdx.x * 256 + threadIdx.x;
    if (k >= KCODES) return;
    const float* e = emb + k * 64;
    float s = 0.f;
    #pragma unroll
    for (int d = 0; d < 64; ++d) s += e[d] * e[d];
    enorm[k] = s;
}

// -------------------- VQ argmin via WMMA f32 16x16x4 -----------------------
// One wave handles 16 positions; loops over all 32 codebook tiles.
// argmin_k(|q|^2 - 2 q.e_k + |e_k|^2) == argmin_k(|e_k|^2 - 2 q.e_k)
__global__ __launch_bounds__(256) void vq_argmin_kernel(
    const float* __restrict__ q,     // [65536, 64] position-major
    const float* __restrict__ emb,   // [512, 64]
    const float* __restrict__ enorm, // [512]
    int* __restrict__ idx)           // [65536]
{
    const int wave = threadIdx.x >> 5;
    const int lane = threadIdx.x & 31;
    const int mtile = blockIdx.x * 8 + wave;   // 4096 tiles total
    const int lg = lane >> 4;
    const int ln = lane & 15;

    const float* arow = q + (mtile * 16 + ln) * 64;

    float minval[8];
    int   minidx[8];
    #pragma unroll
    for (int v = 0; v < 8; ++v) { minval[v] = 3.4e38f; minidx[v] = 0; }

    for (int nt = 0; nt < 32; ++nt) {
        v8f acc = {};
        const float* brow = emb + (nt * 16 + ln) * 64;
        #pragma unroll
        for (int k0 = 0; k0 < 64; k0 += 4) {
            v2f a = *(const v2f*)(arow + k0 + 2 * lg);   // A[ln][k0+2lg+v]
            v2f b = *(const v2f*)(brow + k0 + 2 * lg);   // B[k][ln] = emb[n][k]
            acc = __builtin_amdgcn_wmma_f32_16x16x4_f32(
                false, a, false, b, (short)0, acc, false, false);
        }
        int ncode = nt * 16 + ln;
        float en = enorm[ncode];
        #pragma unroll
        for (int v = 0; v < 8; ++v) {
            float d2 = en - 2.0f * acc[v];               // row m = v + 8*lg
            if (d2 < minval[v]) { minval[v] = d2; minidx[v] = ncode; }
        }
    }
    // reduce across the 16 lanes of each half (xor <= 8 stays in-group)
    #pragma unroll
    for (int off = 8; off >= 1; off >>= 1) {
        #pragma unroll
        for (int v = 0; v < 8; ++v) {
            float ov = __shfl_xor(minval[v], off, 32);
            int   oi = __shfl_xor(minidx[v], off, 32);
            if (ov < minval[v] || (ov == minval[v] && oi < minidx[v])) {
                minval[v] = ov; minidx[v] = oi;
            }
        }
    }
    if (ln == 0) {
        #pragma unroll
        for (int v = 0; v < 8; ++v)
            idx[mtile * 16 + v + 8 * lg] = minidx[v];
    }
}

// -------- build quant (NCHW) from codebook + per-block VQ loss part --------
__global__ __launch_bounds__(256) void quantize_kernel(
    const float* __restrict__ q,   // [65536, 64]
    const float* __restrict__ emb, // [512, 64]
    const int* __restrict__ idx,   // [65536]
    float* __restrict__ quant,     // [1024, 64, 8, 8]
    float* __restrict__ partial)
{
    int t = blockIdx.x * 256 + threadIdx.x;   // VQ_ELEMS exact multiple of 256
    int pos = t >> 6;
    int c   = t & 63;
    int b   = pos >> 6;
    int l   = pos & 63;
    float qo = emb[idx[pos] * 64 + c];
    float qi = q[t];
    quant[(b * 64 + c) * 64 + l] = qo;        // straight-through forward == qo
    float diff = qo - qi;
    float val = diff * diff;

    __shared__ float sred[256];
    sred[threadIdx.x] = val;
    __syncthreads();
    for (int s = 128; s; s >>= 1) {
        if (threadIdx.x < s) sred[threadIdx.x] += sred[threadIdx.x + s];
        __syncthreads();
    }
    if (threadIdx.x == 0) partial[blockIdx.x] = sred[0];
}

// ------------- Q-head layer 1 via WMMA f32 16x16x4 (+bias+ReLU) ------------
// hidden[1024,512] = relu(flat[1024,4096] @ lw1[4096,512] + lb1)
// Block = 8 waves sharing one 32-column slab of lw1; B chunks (64k x 32n,
// 8 KB) are staged into double-buffered LDS with CDNA5 async global->LDS
// copies (ASYNCcnt) and consumed by all waves. One barrier per chunk:
//   wait own ASYNCcnt(c) -> barrier -> issue(c+1) -> compute(c)
__global__ __launch_bounds__(256) void qhead1_kernel(
    const float* __restrict__ flat, const float* __restrict__ lw1,
    const float* __restrict__ lb1, float* __restrict__ hidden)
{
    __shared__ float smem[2][64 * 32];        // [buf][k][n], 2 x 8 KB

    const int wave = threadIdx.x >> 5;
    const int lane = threadIdx.x & 31;
    const int mt = (blockIdx.x >> 4) * 8 + wave;  // 64 m-tiles, 8 per block
    const int np = blockIdx.x & 15;               // 16 column-slabs of 32
    const int lg = lane >> 4;
    const int ln = lane & 15;

    const float* arow = flat + (mt * 16 + ln) * 4096;
    const int ncol0 = np * 32;

    const uint32_t lds0 = (uint32_t)(uintptr_t)(&smem[0][0]);
    const uint32_t lds1 = (uint32_t)(uintptr_t)(&smem[1][0]);

    // stage one 64x32 chunk of lw1 into LDS buf: 512 b128 words, 2 per thread
    auto issue_chunk = [&](int chunk, int buf) {
        uint32_t lb = buf ? lds1 : lds0;
        #pragma unroll
        for (int r = 0; r < 2; ++r) {
            int w  = threadIdx.x + r * 256;
            int k  = w >> 3;                  // 0..63
            int nq = w & 7;                   // 0..7 (quad of columns)
            uint64_t gaddr = (uint64_t)(uintptr_t)
                (lw1 + (chunk * 64 + k) * 512 + ncol0 + nq * 4);
            uint32_t laddr = lb + (uint32_t)(k * 128 + nq * 16);
            asm volatile("global_load_async_to_lds_b128 %0, %1, off"
                         :: "v"(laddr), "v"(gaddr) : "memory");
        }
    };

    v8f acc0 = {}, acc1 = {};
    issue_chunk(0, 0);
    for (int c = 0; c < 64; ++c) {
        asm volatile("s_wait_asynccnt 0x0" ::: "memory");  // my chunk-c done
        __syncthreads();                                   // all chunk-c visible
        if (c + 1 < 64) issue_chunk(c + 1, (c + 1) & 1);   // overlap next copy
        const float* bs = smem[c & 1];
        const float* ap = arow + c * 64;
        #pragma unroll
        for (int kk = 0; kk < 64; kk += 4) {
            const int kb = kk + 2 * lg;
            v2f a = *(const v2f*)(ap + kb);
            v2f b0, b1;
            b0.x = bs[kb * 32 + ln];
            b0.y = bs[(kb + 1) * 32 + ln];
            b1.x = bs[kb * 32 + 16 + ln];
            b1.y = bs[(kb + 1) * 32 + 16 + ln];
            acc0 = __builtin_amdgcn_wmma_f32_16x16x4_f32(
                false, a, false, b0, (short)0, acc0, false, false);
            acc1 = __builtin_amdgcn_wmma_f32_16x16x4_f32(
                false, a, false, b1, (short)0, acc1, false, false);
        }
    }

    const int n0 = ncol0 + ln;
    const int n1 = ncol0 + 16 + ln;
    float bias0 = lb1[n0], bias1 = lb1[n1];
    #pragma unroll
    for (int v = 0; v < 8; ++v) {
        int m = mt * 16 + v + 8 * lg;
        float h0 = acc0[v] + bias0;
        float h1 = acc1[v] + bias1;
        hidden[m * 512 + n0] = h0 > 0.f ? h0 : 0.f;
        hidden[m * 512 + n1] = h1 > 0.f ? h1 : 0.f;
    }
}

// -------------------------- Q-head layer 2 ---------------------------------
__global__ __launch_bounds__(256) void qhead2_kernel(
    const float* __restrict__ hidden, const float* __restrict__ lw2,
    const float* __restrict__ lb2, float* __restrict__ out)
{
    int t = blockIdx.x * 256 + threadIdx.x;
    if (t >= NB * NACT) return;
    int a = t % NACT;
    int b = t / NACT;
    const float* h = hidden + b * 512;
    float acc = lb2[a];
    for (int j = 0; j < 512; ++j)
        acc += h[j] * lw2[j * NACT + a];
    out[t] = acc;
}

// --------------------- deterministic final reduce --------------------------
__global__ void reduce_scale_kernel(const float* __restrict__ part, int n,
                                    float scale, float* __restrict__ dst)
{
    __shared__ float sred[256];
    float s = 0.f;
    for (int i = threadIdx.x; i < n; i += 256) s += part[i];
    sred[threadIdx.x] = s;
    __syncthreads();
    for (int k = 128; k; k >>= 1) {
        if (threadIdx.x < k) sred[threadIdx.x] += sred[threadIdx.x + k];
        __syncthreads();
    }
    if (threadIdx.x == 0) dst[0] = sred[0] * scale;
}

// ---------------------------------------------------------------------------
extern "C" void kernel_launch(void* const* d_in, const int* in_sizes, int n_in,
                              void* d_out, int out_size, void* d_ws, size_t ws_size,
                              hipStream_t stream) {
    const float* x   = (const float*)d_in[0];
    const float* w1  = (const float*)d_in[1];
    const float* b1  = (const float*)d_in[2];
    const float* w2  = (const float*)d_in[3];
    const float* b2  = (const float*)d_in[4];
    const float* w3  = (const float*)d_in[5];
    const float* b3  = (const float*)d_in[6];
    const float* emb = (const float*)d_in[7];
    const float* dw1 = (const float*)d_in[8];
    const float* db1 = (const float*)d_in[9];
    const float* dw2 = (const float*)d_in[10];
    const float* db2 = (const float*)d_in[11];
    const float* dw3 = (const float*)d_in[12];
    const float* db3 = (const float*)d_in[13];
    const float* lw1 = (const float*)d_in[14];
    const float* lb1 = (const float*)d_in[15];
    const float* lw2 = (const float*)d_in[16];
    const float* lb2 = (const float*)d_in[17];
    float* out = (float*)d_out;   // [0..18431] q_values, [18432] rec, [18433] vq

    // ---- workspace layout (floats, 64-float aligned) ----
    float* ws = (float*)d_ws;
    size_t o = 0;
    auto alloc = [&](size_t nf) { float* p = ws + o; o += (nf + 63) & ~(size_t)63; return p; };
    float* h1      = alloc(1024UL * 32 * 21 * 21);   // 14,450,688
    float* h2      = alloc(1024UL * 64 * 10 * 10);   //  6,553,600
    float* qbuf    = alloc((size_t)VQ_ELEMS);        //  4,194,304
    int*   idx     = (int*)alloc(NPOS);              //     65,536
    float* quant   = alloc((size_t)VQ_ELEMS);        //  4,194,304
    float* enorm   = alloc(KCODES);
    float* hidden  = alloc(1024UL * 512);
    float* vq_part = alloc(VQ_ELEMS / 256);          //     16,384
    float* rec_part= alloc((REC_ELEMS + 255) / 256); //    112,896
    float* dbuf1   = h2;   // reuse (same size as h2)
    float* dbuf2   = h1;   // reuse (same size as h1)
    (void)ws_size; (void)in_sizes; (void)n_in; (void)out_size;

    const float enc_scale = 1.0f / (255.0f * 255.0f);   // x/255 then /255 again

    // encoder
    {
        int total = 1024 * 32 * 21 * 21;
        conv_relu_kernel<4, 84, 32, 21, 4, 4, false>
            <<<(total + 255) / 256, 256, 0, stream>>>(x, w1, b1, h1, enc_scale);
    }
    {
        int total = 1024 * 64 * 10 * 10;
        conv_relu_kernel<32, 21, 64, 10, 3, 2, false>
            <<<(total + 255) / 256, 256, 0, stream>>>(h1, w2, b2, h2, 1.0f);
    }
    {
        int total = 1024 * 64 * 8 * 8;
        conv_relu_kernel<64, 10, 64, 8, 3, 1, true>
            <<<(total + 255) / 256, 256, 0, stream>>>(h2, w3, b3, qbuf, 1.0f);
    }

    // VQ lookup
    enorm_kernel<<<2, 256, 0, stream>>>(emb, enorm);
    vq_argmin_kernel<<<(NPOS / 16) / 8, 256, 0, stream>>>(qbuf, emb, enorm, idx);
    quantize_kernel<<<VQ_ELEMS / 256, 256, 0, stream>>>(qbuf, emb, idx, quant, vq_part);

    // decoder
    {
        int total = 1024 * 64 * 10 * 10;
        deconv_relu_kernel<64, 8, 64, 3, 1>
            <<<(total + 255) / 256, 256, 0, stream>>>(quant, dw1, db1, dbuf1);
    }
    {
        int total = 1024 * 32 * 21 * 21;
        deconv_relu_kernel<64, 10, 32, 3, 2>
            <<<(total + 255) / 256, 256, 0, stream>>>(dbuf1, dw2, db2, dbuf2);
    }
    {
        int nblk = (REC_ELEMS + 255) / 256;
        deconv3_loss_kernel<<<nblk, 256, 0, stream>>>(dbuf2, dw3, db3, x, rec_part);
    }

    // Q head on detached quantized features (flat = quant NCHW reshape)
    qhead1_kernel<<<128, 256, 0, stream>>>(quant, lw1, lb1, hidden);
    qhead2_kernel<<<(NB * NACT + 255) / 256, 256, 0, stream>>>(hidden, lw2, lb2, out);

    // losses
    reduce_scale_kernel<<<1, 256, 0, stream>>>(
        rec_part, (REC_ELEMS + 255) / 256, 1.0f / (float)REC_ELEMS, out + NB * NACT);
    // quant_loss = codebook + BETA*commitment = (1+0.25) * mean((qo-qi)^2)
    reduce_scale_kernel<<<1, 256, 0, stream>>>(
        vq_part, VQ_ELEMS / 256, 1.25f / (float)VQ_ELEMS, out + NB * NACT + 1);
}